// CameraPoseAnalyzer_27333171871924
// MI455X (gfx1250) — compile-verified
//
#include <hip/hip_runtime.h>
#include <math.h>

typedef __attribute__((ext_vector_type(2))) float v2f;
typedef __attribute__((ext_vector_type(8))) float v8f;

#define NSEL 64
#define WAVES_PER_BLOCK 8
#define MT_PER_WAVE 4
#define FRAMES_PER_BLOCK (WAVES_PER_BLOCK * MT_PER_WAVE * 16)
#define WS_BFLOATS 512   // 8 arrays * 64 floats for the B-side data

// Scale folding: 0.6*min(2*sqrt(d2),1) == sqrt(med3(1.44*d2, 0, 0.36))
// ws float layout: [0]=-2.88*st_x[64] [1]=-2.88*st_y[64] [2]=-2.88*st_z[64]
//                  [3]= 1.44*||st||^2[64]
//                  [4]=sq0[64] [5]=sq1[64] [6]=sq2[64] [7]=sq3[64]
// bitmap (optional): 1 bit per frame id, at ws + WS_BFLOATS floats.
__global__ void prep_selected_kernel(const float* __restrict__ pose,
                                     const int* __restrict__ sel,
                                     float* __restrict__ ws,
                                     unsigned* bitmap) {
  int j = threadIdx.x;
  if (j < NSEL) {
    int idx = sel[j];
    const float* p = pose + (size_t)idx * 9;
    float x = p[0], y = p[1], z = p[2];
    ws[0 * NSEL + j] = -2.88f * x;
    ws[1 * NSEL + j] = -2.88f * y;
    ws[2 * NSEL + j] = -2.88f * z;
    ws[3 * NSEL + j] = 1.44f * (x * x + y * y + z * z);
    ws[4 * NSEL + j] = p[3];
    ws[5 * NSEL + j] = p[4];
    ws[6 * NSEL + j] = p[5];
    ws[7 * NSEL + j] = p[6];
    if (bitmap) atomicOr(&bitmap[idx >> 5], 1u << (idx & 31));
  }
}

// 16-lane max reduction via DPP butterfly (stays within each half-wave row):
// quad_perm[1,0,3,2]=xor1, quad_perm[2,3,0,1]=xor2, row_half_mirror=xor7
// (== xor4 once quads uniform), row_mirror=xor15 (== xor8 once 8-groups uniform)
__device__ __forceinline__ float dpp_max16(float v) {
  int t;
  t = __builtin_amdgcn_mov_dpp(__float_as_int(v), 0xB1, 0xf, 0xf, true);
  v = fmaxf(v, __int_as_float(t));
  t = __builtin_amdgcn_mov_dpp(__float_as_int(v), 0x4E, 0xf, 0xf, true);
  v = fmaxf(v, __int_as_float(t));
  t = __builtin_amdgcn_mov_dpp(__float_as_int(v), 0x141, 0xf, 0xf, true);
  v = fmaxf(v, __int_as_float(t));
  t = __builtin_amdgcn_mov_dpp(__float_as_int(v), 0x140, 0xf, 0xf, true);
  v = fmaxf(v, __int_as_float(t));
  return v;
}

template <bool USE_BITMAP>
__global__ __launch_bounds__(256)
void pose_sim_wmma_kernel(const float* __restrict__ pose,
                          const int* __restrict__ frame_indices,
                          const int* __restrict__ sel,
                          const float* __restrict__ ws,
                          const unsigned* __restrict__ bitmap,
                          float* __restrict__ out, int nframes) {
  const int lane   = threadIdx.x & 31;
  const int wave   = threadIdx.x >> 5;
  const int lane16 = lane & 15;
  const bool hi    = lane >= 16;
  const int rowOff = hi ? 8 : 0;

  // -------- Build B fragments for the 4 N-tiles (once per wave) --------
  // B (4x16 f32, 2 VGPRs): lanes 0-15 hold K=0 (v0) / K=1 (v1),
  //                        lanes 16-31 hold K=2 (v0) / K=3 (v1).
  // Trans B K=3 row is the constant 1.44 so the WMMA adds 1.44*ctn itself.
  v2f bt[4], bq[4];
  float stnS[4];
#pragma unroll
  for (int t = 0; t < 4; ++t) {
    int col = t * 16 + lane16;
    if (!hi) {
      bt[t][0] = ws[0 * NSEL + col];   // -2.88*st_x  (K=0)
      bt[t][1] = ws[1 * NSEL + col];   // -2.88*st_y  (K=1)
      bq[t][0] = ws[4 * NSEL + col];   // sq0
      bq[t][1] = ws[5 * NSEL + col];   // sq1
    } else {
      bt[t][0] = ws[2 * NSEL + col];   // -2.88*st_z  (K=2)
      bt[t][1] = 1.44f;                // K=3: multiplies A's ctn slot
      bq[t][0] = ws[6 * NSEL + col];   // sq2
      bq[t][1] = ws[7 * NSEL + col];   // sq3
    }
    stnS[t] = ws[3 * NSEL + col];      // 1.44*||st||^2 for this column
  }

  const long long blockBase =
      (long long)blockIdx.x * FRAMES_PER_BLOCK + (long long)wave * (MT_PER_WAVE * 16);

#pragma unroll 1
  for (int mt = 0; mt < MT_PER_WAVE; ++mt) {
    long long base = blockBase + (long long)mt * 16;
    if (base >= nframes) break;                // wave-uniform exit

    // -------- Load A-side data (clamped so EXEC stays full) --------
    int f  = (int)base + lane16;
    int fc = f < nframes ? f : (nframes - 1);
    int fi = frame_indices[fc];                // honor the gather
    const float* p = pose + (size_t)fi * 9;
    float x = p[0], y = p[1], z = p[2];
    float q0 = p[3], q1 = p[4], q2 = p[5], q3 = p[6];
    float ctn = x * x + y * y + z * z;

    // isin(fi, selected): bitmap probe (1 load + shift) or compare fallback
    int matched;
    if (USE_BITMAP) {
      unsigned w = bitmap[fi >> 5];
      matched = (int)((w >> (fi & 31)) & 1u);
    } else {
      matched = 0;
#pragma unroll
      for (int j = 0; j < NSEL; ++j) matched |= (fi == sel[j]) ? 1 : 0;
    }
    unsigned selmask = (unsigned)__ballot(matched != 0);  // bit k: frame base+k

    // A (16x4 f32, 2 VGPRs): lanes 0-15 -> K=0/1, lanes 16-31 -> K=2/3
    // Trans A K=3 slot carries ctn (pairs with the constant-1.44 B row).
    v2f a_t, a_q;
    a_t[0] = hi ? z : x;
    a_t[1] = hi ? ctn : y;
    a_q[0] = hi ? q2 : q0;
    a_q[1] = hi ? q3 : q1;

    float m[8];
#pragma unroll
    for (int r = 0; r < 8; ++r) m[r] = -1e30f;

    // -------- 4 N-tiles x (trans, quat) WMMAs + fused epilogue --------
#pragma unroll
    for (int t = 0; t < 4; ++t) {
      v8f zero8 = {};
      // ct[r] = 1.44*(ctn - 2*ct.st) for row r, column lane16
      v8f ct = __builtin_amdgcn_wmma_f32_16x16x4_f32(
          false, a_t, false, bt[t], (short)0, zero8, false, false);
      v8f cq = __builtin_amdgcn_wmma_f32_16x16x4_f32(
          false, a_q, false, bq[t], (short)0, zero8, false, false);
#pragma unroll
      for (int r = 0; r < 8; ++r) {
        float u   = ct[r] + stnS[t];                        // 1.44*d2
        u         = __builtin_amdgcn_fmed3f(u, 0.0f, 0.36f);
        float sim = fmaf(0.4f, fabsf(cq[r]), __builtin_amdgcn_sqrtf(u));
        m[r] = fmaxf(m[r], sim);
      }
    }

    // -------- max over the 16 columns held across each half-wave --------
#pragma unroll
    for (int r = 0; r < 8; ++r) m[r] = dpp_max16(m[r]);

    // lanes 0 and 16 write rows [base+rowOff .. base+rowOff+7]
    if (lane16 == 0) {
#pragma unroll
      for (int r = 0; r < 8; ++r) {
        long long fr = base + rowOff + r;
        int selbit = (selmask >> (rowOff + r)) & 1;
        if (fr < nframes) out[fr] = selbit ? 0.0f : (1.0f - m[r]);
      }
    }
  }
}

extern "C" void kernel_launch(void* const* d_in, const int* in_sizes, int n_in,
                              void* d_out, int out_size, void* d_ws, size_t ws_size,
                              hipStream_t stream) {
  const float* pose = (const float*)d_in[0];        // (N, 9) f32
  const int* frame_indices = (const int*)d_in[1];   // (N,)   i32
  const int* sel = (const int*)d_in[2];             // (64,)  i32
  float* out = (float*)d_out;
  float* ws  = (float*)d_ws;
  int n = in_sizes[0] / 9;

  size_t bitmapWords = ((size_t)n + 31) / 32;
  size_t need = (size_t)WS_BFLOATS * 4 + bitmapWords * 4;
  bool useBitmap = (ws_size >= need);
  unsigned* bitmap =
      useBitmap ? (unsigned*)((char*)d_ws + (size_t)WS_BFLOATS * 4) : nullptr;

  if (useBitmap) hipMemsetAsync(bitmap, 0, bitmapWords * 4, stream);
  prep_selected_kernel<<<1, NSEL, 0, stream>>>(pose, sel, ws, bitmap);

  int grid = (n + FRAMES_PER_BLOCK - 1) / FRAMES_PER_BLOCK;
  if (useBitmap) {
    pose_sim_wmma_kernel<true><<<grid, 256, 0, stream>>>(
        pose, frame_indices, sel, ws, bitmap, out, n);
  } else {
    pose_sim_wmma_kernel<false><<<grid, 256, 0, stream>>>(
        pose, frame_indices, sel, ws, nullptr, out, n);
  }
}